// TokenReorderer_60936995996121
// MI455X (gfx1250) — compile-verified
//
#include <hip/hip_runtime.h>
#include <hip/hip_bf16.h>
#include <stdint.h>

// ---------------------------------------------------------------------------
// Stable counting sort of 8.4M tokens by 6-bit expert id (MoE token reorder).
// Memory-bound: ~193MB total traffic -> ~8.3us roofline at 23.3 TB/s.
// CDNA5 usage: async global->LDS staging (ASYNCcnt), 320KB LDS per-thread
// private multisplit counters (atomic-free, deterministic, stable),
// b128 LDS ops throughout to keep the LDS phases off the critical path.
// ---------------------------------------------------------------------------

#define NUM_EXPERTS 64
#define TOPK        2
#define T_TOKENS    4194304
#define N_FLAT      (T_TOKENS * TOPK)      // 8388608 flat routing slots
#define THREADS     256
#define ITEMS       16                     // contiguous elements per thread
#define CHUNK       (THREADS * ITEMS)      // 4096 elements per block
#define NB          (N_FLAT / CHUNK)       // 2048 blocks
// LDS staging: pad 4 dwords per 16 dwords -> 16B-aligned b128 groups,
// per-thread read stride = 20 dwords (only 2-way bank conflict).
#define STAGE_DW    (CHUNK + (CHUNK / 16) * 4)   // 5120 dwords = 20KB

static_assert(N_FLAT % CHUNK == 0, "grid must tile exactly");

// Async-copy CHUNK int32s (coalesced, b128 per lane) from global to LDS.
// Inline asm so it compiles on both ROCm 7.2 and the amdgpu-toolchain lane.
__device__ __forceinline__ void async_stage_idx(unsigned lds_base_b,
                                                const unsigned* __restrict__ src,
                                                unsigned tid) {
#pragma unroll
  for (int k = 0; k < ITEMS / 4; ++k) {          // 1024 b128 groups total
    unsigned j     = 4u * (unsigned)(k * THREADS) + 4u * tid;  // dword index
    unsigned phys  = j + ((j >> 4) << 2);        // padded dword index
    unsigned lds_b = lds_base_b + phys * 4u;     // 16B aligned
    unsigned g_b   = j * 4u;                     // byte offset from src
    asm volatile("global_load_async_to_lds_b128 %0, %1, %2 offset:0"
                 :: "v"(lds_b), "v"(g_b), "s"(src)
                 : "memory");
  }
  asm volatile("s_wait_asynccnt 0x0" ::: "memory");
}

// Staged (padded) LDS dword index of block-local element j = t*ITEMS + i.
__device__ __forceinline__ unsigned stage_at(unsigned t, unsigned i) {
  return t * (ITEMS + 4) + i;                    // t*20 + i  (i < 16)
}

// Zero a 32KB u16 counter array with b128 stores (8 per thread).
__device__ __forceinline__ void zero_cnt(unsigned short* cnt, unsigned t) {
  uint4* z = (uint4*)cnt;
  const uint4 zz = {0u, 0u, 0u, 0u};
#pragma unroll
  for (int q = 0; q < (NUM_EXPERTS * THREADS * 2) / 16 / THREADS; ++q)  // 8
    z[(unsigned)q * THREADS + t] = zz;
}

// --------------------------- Kernel 1: histogram ---------------------------
__global__ __launch_bounds__(THREADS)
void k_hist(const int* __restrict__ expert_idx, unsigned* __restrict__ hist) {
  __align__(16) __shared__ unsigned short cnt[NUM_EXPERTS * THREADS]; // 32KB
  __align__(16) __shared__ unsigned stage[STAGE_DW];                  // 20KB
  __shared__ unsigned partial[THREADS];

  const unsigned t = threadIdx.x;
  const unsigned b = blockIdx.x;

  zero_cnt(cnt, t);

  const unsigned* src = (const unsigned*)expert_idx + (size_t)b * CHUNK;
  async_stage_idx((unsigned)(uintptr_t)&stage[0], src, t);
  __syncthreads();

  // Private per-thread histogram over this thread's 16 contiguous elements.
#pragma unroll
  for (int i = 0; i < ITEMS; ++i) {
    unsigned e = stage[stage_at(t, (unsigned)i)] & 63u;
    cnt[e * THREADS + t] = (unsigned short)(cnt[e * THREADS + t] + 1u);
  }
  __syncthreads();

  // Reduce 256 thread-counts per expert: thread t sums a 64-entry quarter
  // (expert e = t>>2, quarter p = t&3) with 8 b128 loads + packed adds.
  {
    const unsigned e = t >> 2, p = t & 3u;
    const uint4* c4 = (const uint4*)cnt + (e * 32u + p * 8u);
    unsigned s = 0;
#pragma unroll
    for (int q = 0; q < 8; ++q) {
      uint4 w = c4[q];
      s += (w.x & 0xffffu) + (w.x >> 16) + (w.y & 0xffffu) + (w.y >> 16) +
           (w.z & 0xffffu) + (w.z >> 16) + (w.w & 0xffffu) + (w.w >> 16);
    }
    partial[t] = s;
  }
  __syncthreads();
  if (t < NUM_EXPERTS) {
    unsigned tot = partial[4u * t] + partial[4u * t + 1u] +
                   partial[4u * t + 2u] + partial[4u * t + 3u];
    hist[(size_t)t * NB + b] = tot;   // expert-major: flat scan == stable bases
  }
}

// ------------------- Kernel 2: flat exclusive scan + counts ----------------
#define SCAN_THREADS 1024
#define M_ENTRIES    (NUM_EXPERTS * NB)            // 131072
#define PER_THREAD   (M_ENTRIES / SCAN_THREADS)    // 128

__global__ __launch_bounds__(SCAN_THREADS)
void k_scan(unsigned* __restrict__ hist, float* __restrict__ counts_out) {
  __shared__ unsigned s[SCAN_THREADS];
  const unsigned t = threadIdx.x;
  uint4* h4 = (uint4*)hist + (size_t)t * (PER_THREAD / 4);

  unsigned sum = 0;
#pragma unroll 4
  for (int q = 0; q < PER_THREAD / 4; ++q) {
    uint4 w = h4[q];
    sum += w.x + w.y + w.z + w.w;
  }
  s[t] = sum;
  __syncthreads();

  // Hillis-Steele inclusive scan over the 1024 thread sums.
  for (unsigned d = 1; d < SCAN_THREADS; d <<= 1) {
    unsigned v = s[t];
    if (t >= d) v += s[t - d];
    __syncthreads();
    s[t] = v;
    __syncthreads();
  }
  unsigned run = (t == 0) ? 0u : s[t - 1];

#pragma unroll 4
  for (int q = 0; q < PER_THREAD / 4; ++q) {
    uint4 w = h4[q];
    uint4 o;
    o.x = run;
    o.y = run + w.x;
    o.z = o.y + w.y;
    o.w = o.z + w.z;
    run = o.w + w.w;
    h4[q] = o;                                    // in-place exclusive scan
  }
  __syncthreads();

  if (t < NUM_EXPERTS) {
    unsigned start = hist[(size_t)t * NB];
    unsigned end   = (t == NUM_EXPERTS - 1) ? (unsigned)N_FLAT
                                            : hist[(size_t)(t + 1) * NB];
    counts_out[t] = (float)(end - start);         // num_tokens_per_expert
  }
}

// --------------------------- Kernel 3: scatter -----------------------------
__global__ __launch_bounds__(THREADS)
void k_scatter(const int* __restrict__ expert_idx,
               const float* __restrict__ scores,
               const unsigned* __restrict__ hist,
               float* __restrict__ out_scores,
               float* __restrict__ out_index) {
  __align__(16) __shared__ unsigned short loc[NUM_EXPERTS * THREADS]; // 32KB
  __shared__ unsigned baseo[NUM_EXPERTS];          // global base per expert
  __shared__ unsigned partial[THREADS];            // segment totals
  __align__(16) __shared__ unsigned stage[STAGE_DW];

  const unsigned t = threadIdx.x;
  const unsigned b = blockIdx.x;

  zero_cnt(loc, t);

  const unsigned* src = (const unsigned*)expert_idx + (size_t)b * CHUNK;
  const float*    sc  = scores + (size_t)b * CHUNK;
  __builtin_prefetch(sc + (size_t)t * ITEMS, 0, 0);       // global_prefetch_b8

  async_stage_idx((unsigned)(uintptr_t)&stage[0], src, t);
  if (t < NUM_EXPERTS) baseo[t] = hist[(size_t)t * NB + b];
  __syncthreads();

  // Pass A: private counts (stable (thread,item) order == element order).
#pragma unroll
  for (int i = 0; i < ITEMS; ++i) {
    unsigned e = stage[stage_at(t, (unsigned)i)] & 63u;
    loc[e * THREADS + t] = (unsigned short)(loc[e * THREADS + t] + 1u);
  }
  __syncthreads();

  // Two-level exclusive scan across threads per expert, all 256 threads busy.
  // Thread t owns segment (expert e = t>>2, quarter p = t&3): 64 u16 counts
  // = 8 b128 loads kept in registers as 32 packed words.
  const unsigned e = t >> 2, p = t & 3u;
  uint4* c4 = (uint4*)loc + (e * 32u + p * 8u);
  uint4 w[8];
  unsigned segtot = 0;
#pragma unroll
  for (int q = 0; q < 8; ++q) {
    w[q] = c4[q];
    segtot += (w[q].x & 0xffffu) + (w[q].x >> 16) +
              (w[q].y & 0xffffu) + (w[q].y >> 16) +
              (w[q].z & 0xffffu) + (w[q].z >> 16) +
              (w[q].w & 0xffffu) + (w[q].w >> 16);
  }
  partial[t] = segtot;
  __syncthreads();

  // Segment base = sum of lower quarters of the same expert (thread order).
  {
    unsigned run = 0;
    if (p > 0u) run += partial[t - p + 0u];
    if (p > 1u) run += partial[t - p + 1u];
    if (p > 2u) run += partial[t - p + 2u];
    // Exclusive prefix over the 64 entries, computed in registers on packed
    // u16 pairs, then stored back with b128 stores.
#pragma unroll
    for (int q = 0; q < 8; ++q) {
      uint4 v = w[q], o;
      unsigned lo, hi;
      lo = v.x & 0xffffu; hi = v.x >> 16;
      o.x = run | ((run + lo) << 16); run += lo + hi;
      lo = v.y & 0xffffu; hi = v.y >> 16;
      o.y = run | ((run + lo) << 16); run += lo + hi;
      lo = v.z & 0xffffu; hi = v.z >> 16;
      o.z = run | ((run + lo) << 16); run += lo + hi;
      lo = v.w & 0xffffu; hi = v.w >> 16;
      o.w = run | ((run + lo) << 16); run += lo + hi;
      c4[q] = o;
    }
  }
  __syncthreads();

  // Preload this thread's 16 scores with b128 loads (coalesced cachelines).
  float4 vv[ITEMS / 4];
  const float4* scv = (const float4*)(sc + (size_t)t * ITEMS);
#pragma unroll
  for (int q = 0; q < ITEMS / 4; ++q) vv[q] = scv[q];
  const float* v = (const float*)&vv[0];

  // Pass B: in-order emit; LDS slot is thread-private -> no atomics needed.
#pragma unroll
  for (int i = 0; i < ITEMS; ++i) {
    unsigned j   = t * ITEMS + (unsigned)i;
    unsigned ee  = stage[stage_at(t, (unsigned)i)] & 63u;
    unsigned r   = loc[ee * THREADS + t];
    loc[ee * THREADS + t] = (unsigned short)(r + 1u);
    unsigned pos = baseo[ee] + r;
    unsigned gj  = b * CHUNK + j;
    out_scores[pos] = v[i];
    out_index[pos]  = (float)gj;    // < 2^23 -> exact in f32
  }
}

// ------------------------------- launcher ----------------------------------
extern "C" void kernel_launch(void* const* d_in, const int* in_sizes, int n_in,
                              void* d_out, int out_size, void* d_ws, size_t ws_size,
                              hipStream_t stream) {
  (void)in_sizes; (void)n_in; (void)out_size; (void)ws_size;
  const float* top_scores = (const float*)d_in[0];
  const int*   sel        = (const int*)d_in[1];   // int32 per harness convention

  float* out        = (float*)d_out;
  float* out_scores = out;                          // [0, N)
  float* out_index  = out + (size_t)N_FLAT;         // [N, 2N)
  float* out_counts = out + 2 * (size_t)N_FLAT;     // [2N, 2N+64)

  unsigned* hist = (unsigned*)d_ws;                 // 131072 u32 = 512KB scratch

  k_hist   <<<NB, THREADS,      0, stream>>>(sel, hist);
  k_scan   <<<1,  SCAN_THREADS, 0, stream>>>(hist, out_counts);
  k_scatter<<<NB, THREADS,      0, stream>>>(sel, top_scores, hist,
                                             out_scores, out_index);
}